// PackerProteinFeatures_84361747628659
// MI455X (gfx1250) — compile-verified
//
#include <hip/hip_runtime.h>
#include <math.h>

// ---------------- constants ----------------
constexpr int LL    = 2048;
constexpr int MM    = 16;
constexpr int TOPK  = 30;
constexpr int NF    = 128;   // NODE_F == EDGE_F

typedef __attribute__((ext_vector_type(2))) float v2f;
typedef __attribute__((ext_vector_type(8))) float v8f;

__device__ __constant__ int GROUP_D[119] = {
0,1,18,1,2,13,14,15,16,17,18,1,2,13,14,15,16,17,18,1,2,3,4,5,6,7,8,9,10,11,12,13,
14,15,16,17,18,1,2,3,4,5,6,7,8,9,10,11,12,13,14,15,16,17,18,1,2,3,3,3,3,3,3,3,3,3,
3,3,3,3,3,3,4,5,6,7,8,9,10,11,12,13,14,15,16,17,18,1,2,3,3,3,3,3,3,3,3,3,3,3,3,3,
3,3,4,5,6,7,8,9,10,11,12,13,14,15,16,17,18};
__device__ __constant__ int PERIOD_D[119] = {
0,1,1,2,2,2,2,2,2,2,2,3,3,3,3,3,3,3,3,4,4,4,4,4,4,4,4,4,4,4,4,4,4,4,4,4,4,5,5,5,
5,5,5,5,5,5,5,5,5,5,5,5,5,5,5,6,6,6,6,6,6,6,6,6,6,6,6,6,6,6,6,6,6,6,6,6,6,6,6,6,
6,6,6,6,6,6,6,7,7,7,7,7,7,7,7,7,7,7,7,7,7,7,7,7,7,7,7,7,7,7,7,7,7,7,7,7,7,7,7};

// rbf(d) with mu = linspace(0,20,16), sigma = 20/16
__device__ __forceinline__ float rbf_val(float d, int t) {
  float mu = (20.0f / 15.0f) * (float)t;
  float z  = (d - mu) * (1.0f / 1.25f);
  return __expf(-z * z);
}

// ---------------- kernel 1: per-residue geometry ----------------
// coords layout [5][LL][3]: n, ca, c, o, cb ; frames [LL][9]: e1,e2,e3
__global__ void geo_kernel(const float* __restrict__ x,
                           float* __restrict__ coords,
                           float* __restrict__ frames) {
  int l = blockIdx.x * blockDim.x + threadIdx.x;
  if (l >= LL) return;
  const float* p = x + (size_t)l * 12;
  float n0=p[0],n1=p[1],n2=p[2], a0=p[3],a1=p[4],a2=p[5];
  float c0=p[6],c1=p[7],c2=p[8], o0=p[9],o1=p[10],o2=p[11];
  // cb
  float b0=a0-n0, b1=a1-n1, b2=a2-n2;
  float v0=c0-a0, v1=c1-a1, v2=c2-a2;
  float x0=b1*v2-b2*v1, x1=b2*v0-b0*v2, x2=b0*v1-b1*v0;
  float cb0=-0.58273431f*x0+0.56802827f*b0-0.54067466f*v0+a0;
  float cb1=-0.58273431f*x1+0.56802827f*b1-0.54067466f*v1+a1;
  float cb2=-0.58273431f*x2+0.56802827f*b2-0.54067466f*v2+a2;
  float* c_n  = coords + (size_t)(0*LL + l)*3;
  float* c_ca = coords + (size_t)(1*LL + l)*3;
  float* c_c  = coords + (size_t)(2*LL + l)*3;
  float* c_o  = coords + (size_t)(3*LL + l)*3;
  float* c_cb = coords + (size_t)(4*LL + l)*3;
  c_n[0]=n0;  c_n[1]=n1;  c_n[2]=n2;
  c_ca[0]=a0; c_ca[1]=a1; c_ca[2]=a2;
  c_c[0]=c0;  c_c[1]=c1;  c_c[2]=c2;
  c_o[0]=o0;  c_o[1]=o1;  c_o[2]=o2;
  c_cb[0]=cb0;c_cb[1]=cb1;c_cb[2]=cb2;
  // frame: v1 = n-ca, v2 = c-ca
  float w0=n0-a0, w1=n1-a1, w2=n2-a2;
  float u0=c0-a0, u1=c1-a1, u2=c2-a2;
  float inl = 1.0f/(sqrtf(w0*w0+w1*w1+w2*w2)+1e-8f);
  float e10=w0*inl, e11=w1*inl, e12=w2*inl;
  float d = e10*u0+e11*u1+e12*u2;
  float q0=u0-e10*d, q1=u1-e11*d, q2=u2-e12*d;
  float inq = 1.0f/(sqrtf(q0*q0+q1*q1+q2*q2)+1e-8f);
  float e20=q0*inq, e21=q1*inq, e22=q2*inq;
  float e30=e11*e22-e12*e21, e31=e12*e20-e10*e22, e32=e10*e21-e11*e20;
  float* fr = frames + (size_t)l*9;
  fr[0]=e10; fr[1]=e11; fr[2]=e12;
  fr[3]=e20; fr[4]=e21; fr[5]=e22;
  fr[6]=e30; fr[7]=e31; fr[8]=e32;
}

// ---------------- kernel 2: top-K neighbors ----------------
__global__ __launch_bounds__(256)
void topk_kernel(const float* __restrict__ x, const float* __restrict__ mask,
                 int* __restrict__ eidx_ws, float* __restrict__ eidx_out) {
  __shared__ float d2s[LL];
  __shared__ float rv[256];
  __shared__ int   ri[256];
  int i = blockIdx.x, tid = threadIdx.x;
  const float* pa = x + (size_t)i*12 + 3;
  float ax=pa[0], ay=pa[1], az=pa[2];
  float mi = mask[i];
  for (int j = tid; j < LL; j += 256) {
    const float* pb = x + (size_t)j*12 + 3;
    float dx=ax-pb[0], dy=ay-pb[1], dz=az-pb[2];
    float d2 = dx*dx+dy*dy+dz*dz;
    d2s[j] = (mi * mask[j] > 0.0f) ? d2 : 1e8f;
  }
  __syncthreads();
  for (int k = 0; k < TOPK; ++k) {
    float bv = 3.0e38f; int bi = LL;
    for (int j = tid; j < LL; j += 256) {
      float v = d2s[j];
      if (v < bv) { bv = v; bi = j; }
    }
    rv[tid] = bv; ri[tid] = bi;
    __syncthreads();
    for (int s = 128; s > 0; s >>= 1) {
      if (tid < s) {
        float v2 = rv[tid+s]; int i2 = ri[tid+s];
        if (v2 < rv[tid] || (v2 == rv[tid] && i2 < ri[tid])) { rv[tid]=v2; ri[tid]=i2; }
      }
      __syncthreads();
    }
    if (tid == 0) {
      int w = ri[0];
      eidx_ws[i*TOPK + k]  = w;
      eidx_out[i*TOPK + k] = (float)w;
      d2s[w] = 3.0e38f;
    }
    __syncthreads();
  }
}

// ---------------- generic gather + layernorm (128-wide rows) ----------------
__device__ __forceinline__ void row_ln_store(float val, int c, const float* g,
                                             const float* b, float* out, float* red) {
  red[c] = val; __syncthreads();
  for (int s = 64; s > 0; s >>= 1) { if (c < s) red[c] += red[c+s]; __syncthreads(); }
  float mu = red[0] * (1.0f/128.0f);
  __syncthreads();
  float d = val - mu;
  red[c] = d * d; __syncthreads();
  for (int s = 64; s > 0; s >>= 1) { if (c < s) red[c] += red[c+s]; __syncthreads(); }
  float rs = rsqrtf(red[0] * (1.0f/128.0f) + 1e-5f);
  out[c] = d * rs * g[c] + b[c];
}

__global__ __launch_bounds__(128)
void vseq_kernel(const int* __restrict__ s, const float* __restrict__ W,
                 const float* __restrict__ g, const float* __restrict__ b,
                 float* __restrict__ out) {
  __shared__ float red[128];
  int l = blockIdx.x, c = threadIdx.x;
  float val = W[(size_t)s[l]*NF + c];
  row_ln_store(val, c, g, b, out + (size_t)l*NF, red);
}

__global__ __launch_bounds__(128)
void ynodes_kernel(const int* __restrict__ y_t, const float* __restrict__ W,
                   const float* __restrict__ g, const float* __restrict__ b,
                   float* __restrict__ out) {
  __shared__ float red[128];
  int row = blockIdx.x, c = threadIdx.x;
  int tt = y_t[row];
  float val = W[(size_t)tt*NF + c]
            + W[(size_t)(120 + GROUP_D[tt])*NF + c]
            + W[(size_t)(139 + PERIOD_D[tt])*NF + c];
  row_ln_store(val, c, g, b, out + (size_t)row*NF, red);
}

// ---------------- WMMA f32 16x16x4 tile GEMM helper ----------------
// A (16xK) in LDS (row stride FS), B (Kx128) in global. Wave w owns cols [16w,16w+16).
__device__ __forceinline__ v8f wmma_rows_x_weights(const float* featLDS, int FS, int K,
                                                   const float* __restrict__ W,
                                                   int lane, int col) {
  int half = lane >> 4;
  int l16  = lane & 15;
  v8f acc = {};
  for (int kk = 0; kk < K; kk += 4) {
    v2f a, b;
    a.x = featLDS[l16 * FS + kk + 2*half + 0];
    a.y = featLDS[l16 * FS + kk + 2*half + 1];
    b.x = W[(size_t)(kk + 2*half + 0) * NF + col];
    b.y = W[(size_t)(kk + 2*half + 1) * NF + col];
    acc = __builtin_amdgcn_wmma_f32_16x16x4_f32(false, a, false, b, (short)0, acc,
                                                false, false);
  }
  return acc;
}

// layernorm over 16x128 LDS tile, then store with gamma/beta
__device__ __forceinline__ void tile_ln_store(float* tile, float* mu_s, float* rs_s,
                                              const float* __restrict__ g,
                                              const float* __restrict__ b,
                                              float* __restrict__ out, int tid) {
  if (tid < 16) {
    float s = 0.f;
    for (int c = 0; c < NF; ++c) s += tile[tid*NF + c];
    float m = s * (1.0f/128.0f);
    float v = 0.f;
    for (int c = 0; c < NF; ++c) { float d = tile[tid*NF + c] - m; v += d*d; }
    mu_s[tid] = m;
    rs_s[tid] = rsqrtf(v * (1.0f/128.0f) + 1e-5f);
  }
  __syncthreads();
  for (int idx = tid; idx < 16*NF; idx += 256) {
    int rr = idx >> 7, cc = idx & 127;
    out[(size_t)rr*NF + cc] = (tile[idx] - mu_s[rr]) * rs_s[rr] * g[cc] + b[cc];
  }
}

// ---------------- kernel 3: edge features + GEMM(416x128) + LN ----------------
constexpr int KE  = 416;
constexpr int FSE = 417;   // LDS row stride, gcd(417,64)==1 -> bank-conflict free
__global__ __launch_bounds__(256)
void edge_kernel(const float* __restrict__ coords, const int* __restrict__ eidx,
                 const int* __restrict__ r_idx, const int* __restrict__ chains,
                 const float* __restrict__ pos_w, const float* __restrict__ pos_b,
                 const float* __restrict__ W, const float* __restrict__ g,
                 const float* __restrict__ b, float* __restrict__ out) {
  __shared__ float feat[16 * FSE];
  __shared__ float tile[16 * NF];
  __shared__ float mu_s[16], rs_s[16];
  int tid = threadIdx.x;
  int tb  = blockIdx.x * 16;          // first edge row of tile
  {
    int r = tid >> 4, t = tid & 15;
    int E = tb + r;
    int i = E / TOPK, k = E % TOPK; (void)k;
    int j = eidx[E];
    int off = r_idx[i] - r_idx[j];
    int ech = (chains[i] == chains[j]);
    int bin = ech ? min(max(off + 32, 0), 64) : 65;
    feat[r*FSE + t] = pos_w[bin*16 + t] + pos_b[t];
    for (int p = 0; p < 25; ++p) {
      int ai = p / 5, aj = p % 5;
      const float* ci = coords + (size_t)(ai*LL + i)*3;
      const float* cj = coords + (size_t)(aj*LL + j)*3;
      float dx=ci[0]-cj[0], dy=ci[1]-cj[1], dz=ci[2]-cj[2];
      float d = sqrtf(dx*dx + dy*dy + dz*dz + 1e-6f);
      feat[r*FSE + 16 + p*16 + t] = rbf_val(d, t);
    }
  }
  __syncthreads();
  int wave = tid >> 5, lane = tid & 31;
  int col  = wave*16 + (lane & 15);
  int half = lane >> 4;
  v8f acc = wmma_rows_x_weights(feat, FSE, KE, W, lane, col);
  for (int q = 0; q < 8; ++q) tile[(q + 8*half)*NF + col] = acc[q];
  __syncthreads();
  tile_ln_store(tile, mu_s, rs_s, g, b, out + (size_t)tb*NF, tid);
}

// ---------------- kernel 4: per-(l,m) features + GEMM(148x128)+bias + LN ------
constexpr int KV  = 148;
constexpr int FSV = 149;
__global__ __launch_bounds__(256)
void vnode_kernel(const float* __restrict__ coords, const float* __restrict__ frames,
                  const float* __restrict__ y, const int* __restrict__ y_t,
                  const float* __restrict__ type_w, const float* __restrict__ type_b,
                  const float* __restrict__ W, const float* __restrict__ bias,
                  const float* __restrict__ g, const float* __restrict__ b,
                  float* __restrict__ out) {
  __shared__ float feat[16 * FSV];
  __shared__ float tile[16 * NF];
  __shared__ float mu_s[16], rs_s[16];
  int tid = threadIdx.x;
  int tb  = blockIdx.x * 16;
  {
    int r = tid >> 4, t = tid & 15;
    int E = tb + r;
    int l = E >> 4, m = E & 15;
    const float* yp = y + (size_t)(l*MM + m)*3;
    float yx = yp[0], yy = yp[1], yz = yp[2];
    for (int a = 0; a < 5; ++a) {
      const float* ci = coords + (size_t)(a*LL + l)*3;
      float dx=ci[0]-yx, dy=ci[1]-yy, dz=ci[2]-yz;
      float d = sqrtf(dx*dx + dy*dy + dz*dz + 1e-6f);
      feat[r*FSV + a*16 + t] = rbf_val(d, t);
    }
    int tt = y_t[l*MM + m];
    int gg = GROUP_D[tt], pp = PERIOD_D[tt];
    for (int q = 0; q < 4; ++q) {
      int c = t*4 + q;
      feat[r*FSV + 80 + c] = type_w[tt*64 + c] + type_w[(120+gg)*64 + c]
                           + type_w[(139+pp)*64 + c] + type_b[c];
    }
    if (t == 0) {
      const float* fr = frames + (size_t)l*9;
      const float* ca = coords + (size_t)(1*LL + l)*3;
      float rx = yx-ca[0], ry = yy-ca[1], rz = yz-ca[2];
      float lx = fr[0]*rx + fr[1]*ry + fr[2]*rz;
      float ly = fr[3]*rx + fr[4]*ry + fr[5]*rz;
      float lz = fr[6]*rx + fr[7]*ry + fr[8]*rz;
      float rxy  = sqrtf(lx*lx + ly*ly + 1e-8f);
      float rxyz = sqrtf(lx*lx + ly*ly + lz*lz) + 1e-8f;
      feat[r*FSV + 144] = lx / rxy;
      feat[r*FSV + 145] = ly / rxy;
      feat[r*FSV + 146] = rxy / rxyz;
      feat[r*FSV + 147] = lz / rxyz;
    }
  }
  __syncthreads();
  int wave = tid >> 5, lane = tid & 31;
  int col  = wave*16 + (lane & 15);
  int half = lane >> 4;
  v8f acc = wmma_rows_x_weights(feat, FSV, KV, W, lane, col);
  float bb = bias[col];
  for (int q = 0; q < 8; ++q) tile[(q + 8*half)*NF + col] = acc[q] + bb;
  __syncthreads();
  tile_ln_store(tile, mu_s, rs_s, g, b, out + (size_t)tb*NF, tid);
}

// ---------------- kernel 5: y_edges rbf GEMM(16x128) + LN ----------------
constexpr int KY  = 16;
constexpr int FSY = 17;
__global__ __launch_bounds__(256)
void yedge_kernel(const float* __restrict__ y, const float* __restrict__ W,
                  const float* __restrict__ g, const float* __restrict__ b,
                  float* __restrict__ out) {
  __shared__ float feat[16 * FSY];
  __shared__ float tile[16 * NF];
  __shared__ float mu_s[16], rs_s[16];
  int tid = threadIdx.x;
  int tb  = blockIdx.x * 16;
  {
    int r = tid >> 4, t = tid & 15;
    int E = tb + r;
    int l = E >> 8, rem = E & 255;
    int m1 = rem >> 4, m2 = rem & 15;
    const float* y1 = y + (size_t)(l*MM + m1)*3;
    const float* y2 = y + (size_t)(l*MM + m2)*3;
    float dx=y1[0]-y2[0], dy=y1[1]-y2[1], dz=y1[2]-y2[2];
    float d = sqrtf(dx*dx + dy*dy + dz*dz + 1e-6f);
    feat[r*FSY + t] = rbf_val(d, t);
  }
  __syncthreads();
  int wave = tid >> 5, lane = tid & 31;
  int col  = wave*16 + (lane & 15);
  int half = lane >> 4;
  v8f acc = wmma_rows_x_weights(feat, FSY, KY, W, lane, col);
  for (int q = 0; q < 8; ++q) tile[(q + 8*half)*NF + col] = acc[q];
  __syncthreads();
  tile_ln_store(tile, mu_s, rs_s, g, b, out + (size_t)tb*NF, tid);
}

// ---------------- launcher ----------------
extern "C" void kernel_launch(void* const* d_in, const int* in_sizes, int n_in,
                              void* d_out, int out_size, void* d_ws, size_t ws_size,
                              hipStream_t stream) {
  const int*   s       = (const int*)  d_in[0];
  const float* x       = (const float*)d_in[1];
  const float* y       = (const float*)d_in[2];
  // d_in[3] = y_m (unused by reference)
  const int*   y_t     = (const int*)  d_in[4];
  const float* mask    = (const float*)d_in[5];
  const int*   r_idx   = (const int*)  d_in[6];
  const int*   chains  = (const int*)  d_in[7];
  const float* pos_w       = (const float*)d_in[8];
  const float* pos_b       = (const float*)d_in[9];
  const float* enc_edge_w  = (const float*)d_in[10];
  const float* enc_ne_g    = (const float*)d_in[11];
  const float* enc_ne_b    = (const float*)d_in[12];
  const float* enc_node_w  = (const float*)d_in[13];
  const float* enc_nn_g    = (const float*)d_in[14];
  const float* enc_nn_b    = (const float*)d_in[15];
  const float* type_w      = (const float*)d_in[16];
  const float* type_b      = (const float*)d_in[17];
  const float* proj_w      = (const float*)d_in[18];
  const float* proj_b      = (const float*)d_in[19];
  const float* nn_g        = (const float*)d_in[20];
  const float* nn_b        = (const float*)d_in[21];
  const float* yn_w        = (const float*)d_in[22];
  const float* ye_w        = (const float*)d_in[23];
  const float* nye_g       = (const float*)d_in[24];
  const float* nye_b       = (const float*)d_in[25];
  const float* nyn_g       = (const float*)d_in[26];
  const float* nyn_b       = (const float*)d_in[27];

  float* out      = (float*)d_out;
  float* out_vseq = out;                                    // (2048,128)
  float* out_e    = out_vseq + (size_t)LL*NF;               // (2048,30,128)
  float* out_eidx = out_e + (size_t)LL*TOPK*NF;             // (2048,30)
  float* out_v    = out_eidx + (size_t)LL*TOPK;             // (2048,128)
  float* out_yn   = out_v + (size_t)LL*NF;                  // (2048,16,128)
  float* out_ye   = out_yn + (size_t)LL*MM*NF;              // (2048,16,16,128)

  float* coords  = (float*)d_ws;                            // 5*2048*3 floats
  float* frames  = coords + (size_t)5*LL*3;                 // 2048*9 floats
  int*   eidx_ws = (int*)(frames + (size_t)LL*9);           // 2048*30 ints

  geo_kernel  <<<(LL + 255)/256, 256, 0, stream>>>(x, coords, frames);
  topk_kernel <<<LL, 256, 0, stream>>>(x, mask, eidx_ws, out_eidx);
  vseq_kernel <<<LL, 128, 0, stream>>>(s, enc_node_w, enc_nn_g, enc_nn_b, out_vseq);
  edge_kernel <<<(LL*TOPK)/16, 256, 0, stream>>>(coords, eidx_ws, r_idx, chains,
                pos_w, pos_b, enc_edge_w, enc_ne_g, enc_ne_b, out_e);
  vnode_kernel<<<(LL*MM)/16, 256, 0, stream>>>(coords, frames, y, y_t,
                type_w, type_b, proj_w, proj_b, nn_g, nn_b, out_v);
  ynodes_kernel<<<LL*MM, 128, 0, stream>>>(y_t, yn_w, nyn_g, nyn_b, out_yn);
  yedge_kernel<<<(LL*MM*MM)/16, 256, 0, stream>>>(y, ye_w, nye_g, nye_b, out_ye);
}